// DecoderStepLayer_77841987273480
// MI455X (gfx1250) — compile-verified
//
#include <hip/hip_runtime.h>
#include <hip/hip_bf16.h>

// Problem dims (fixed by the reference)
#define B_   16
#define LPC  2047   // history length (self-attn before concat)
#define LSC  2048   // self-attn total length (LPC + 1)
#define LEC  2048   // encoder length
#define D_   1024
#define H_   16
#define DK_  64
#define DV_  64
#define DF_  4096

typedef __attribute__((ext_vector_type(2))) float v2f;
typedef __attribute__((ext_vector_type(8))) float v8f;
typedef int v4i_t __attribute__((vector_size(4 * sizeof(int))));

// ---------------------------------------------------------------------------
// fp32 WMMA: D(16x16,f32) = A(16x4,f32) * B(4x16,f32) + C
// A frag: lane m=(lane&15) holds row m, k-offsets kb,kb+1 with kb=(lane>>4)*2
// B frag: lane n=(lane&15) holds col n, k-offsets kb,kb+1
// C/D:    VGPR r -> M=r (lanes 0-15), M=r+8 (lanes 16-31); N=lane&15
// ---------------------------------------------------------------------------
__device__ __forceinline__ v8f mma4(v2f a, v2f b, v8f c) {
  return __builtin_amdgcn_wmma_f32_16x16x4_f32(false, a, false, b,
                                               (short)0, c, false, false);
}

// ---------------------------------------------------------------------------
// Async global->LDS copy (CDNA5 GLOBAL_LOAD_ASYNC_TO_LDS_B128, ASYNCcnt).
// Param 0 is v4i addrspace(1)* (global src), param 1 v4i addrspace(3)* (LDS
// dst), then imm offset / imm cpol. Guarded fallback to direct LDS stores.
// ---------------------------------------------------------------------------
#if __has_builtin(__builtin_amdgcn_global_load_async_to_lds_b128) && \
    __has_builtin(__builtin_amdgcn_s_wait_asynccnt)
#define ASYNC_LDS 1
#else
#define ASYNC_LDS 0
#endif

__device__ __forceinline__ void async_cp16(float* lds_dst, const float* g_src) {
#if ASYNC_LDS
  __builtin_amdgcn_global_load_async_to_lds_b128(
      (__attribute__((address_space(1))) v4i_t*)g_src,
      (__attribute__((address_space(3))) v4i_t*)lds_dst, 0, 0);
#else
  const float4 v = *(const float4*)g_src;
  *(float4*)lds_dst = v;
#endif
}

#if ASYNC_LDS
#define ASYNC_WAIT(n) __builtin_amdgcn_s_wait_asynccnt(n)
#else
#define ASYNC_WAIT(n)
#endif

// LDS tile geometry for the streaming kernels: 64 rows x 64 floats,
// padded to 68 floats/row so 16 lanes reading 16 different rows at the
// same column hit 16 distinct banks (68 % 64 == 4).
#define ROWS_   64
#define KC_     64
#define STRIDE_ 68
#define TILEF_  (ROWS_ * STRIDE_)

// ---------------------------------------------------------------------------
// Generic skinny GEMM: OUT[16, Ntiles*64] = A[16,K] @ W[K,N] (+bias, +relu)
// One wave per 16-column tile, 4 waves per block. blockIdx.y selects a slice
// (e.g. per-head weight matrix) via the sA/sW/sO strides.
// WT=1: W stored N-major (W[n*ldw + k]) -- used for transposed wk access.
// ---------------------------------------------------------------------------
template <int WT, int RELU>
__global__ __launch_bounds__(128) void gemm16_kernel(
    const float* __restrict__ A, int lda, long sA,
    const float* __restrict__ W, int ldw, long sW,
    const float* __restrict__ bias, long sB,
    float* __restrict__ OUT, int ldo, long sO, int K) {
  const int slice = blockIdx.y;
  A += (long)slice * sA;
  W += (long)slice * sW;
  OUT += (long)slice * sO;
  const int lane = threadIdx.x & 31;
  const int wave = threadIdx.x >> 5;
  const int n0 = (blockIdx.x * 4 + wave) * 16;
  const int m = lane & 15;
  const int kb = (lane >> 4) << 1;
  const float* arow = A + (long)m * lda;
  v8f c = {};
  for (int k = 0; k < K; k += 4) {
    v2f a;
    a.x = arow[k + kb];
    a.y = arow[k + kb + 1];
    v2f b;
    if (WT) {
      const float* wrow = W + (long)(n0 + m) * ldw;
      b.x = wrow[k + kb];
      b.y = wrow[k + kb + 1];
    } else {
      if ((k & 15) == 0)
        __builtin_prefetch(&W[(long)(k + 16) * ldw + n0 + m], 0, 1);
      b.x = W[(long)(k + kb) * ldw + n0 + m];
      b.y = W[(long)(k + kb + 1) * ldw + n0 + m];
    }
    c = mma4(a, b, c);
  }
  const int rb = (lane >> 4) * 8;
  const float bv = bias ? bias[(long)slice * sB + n0 + m] : 0.0f;
#pragma unroll
  for (int r = 0; r < 8; ++r) {
    float v = c[r] + bv;
    if (RELU) v = fmaxf(v, 0.0f);
    OUT[(long)(r + rb) * ldo + n0 + m] = v;
  }
}

// ---------------------------------------------------------------------------
// Fused attention scores: logits[b,h,l] = (qw[b,h,:] . kv[b,l,:]) * scale
// kv row l: hist for l < LP, else the appended token row (last).
// Block tiles 64 l-rows; kv staged through double-buffered LDS via async
// b128 copies; B fragments served from LDS, A (qw) streamed from L2.
// ---------------------------------------------------------------------------
__global__ __launch_bounds__(128) void scores_kernel(
    const float* __restrict__ qw,    // [B,H,D]
    const float* __restrict__ hist,  // [B,LP,D]
    const float* __restrict__ last,  // [B,D]
    float* __restrict__ logits,      // [B,H,L]
    int LP, int L, float scale) {
  __shared__ float tile[2 * TILEF_];
  const int b = blockIdx.y;
  const int t = threadIdx.x;
  const int lane = t & 31;
  const int wave = t >> 5;
  const int lblk = blockIdx.x * 64;
  const int m = lane & 15;
  const int kb = (lane >> 4) << 1;

  // cooperative loader mapping: 2 threads per kv row, 32 floats each
  const int lrow = t >> 1;
  const int lcol = (t & 1) * 32;
  const int lglob = lblk + lrow;
  const float* grow =
      (lglob < LP) ? (hist + ((long)b * LP + lglob) * D_) : (last + (long)b * D_);

  const float* arow = qw + ((long)b * H_ + m) * D_;
  const int lloc = wave * 16 + m;  // tile row this lane's B column uses

  // preload chunk 0
  {
    float* dst = &tile[lrow * STRIDE_ + lcol];
#pragma unroll
    for (int i = 0; i < 8; ++i) async_cp16(dst + i * 4, grow + lcol + i * 4);
  }

  v8f c = {};
  for (int kc = 0; kc < D_; kc += KC_) {
    const int cur = (kc / KC_) & 1;
    if (kc + KC_ < D_) {
      float* dst = &tile[(cur ^ 1) * TILEF_ + lrow * STRIDE_ + lcol];
      const float* src = grow + kc + KC_ + lcol;
#pragma unroll
      for (int i = 0; i < 8; ++i) async_cp16(dst + i * 4, src + i * 4);
      ASYNC_WAIT(8);
    } else {
      ASYNC_WAIT(0);
    }
    __syncthreads();
    const float* bt = &tile[cur * TILEF_ + lloc * STRIDE_];
#pragma unroll
    for (int kk = 0; kk < KC_; kk += 4) {
      v2f a;
      a.x = arow[kc + kk + kb];
      a.y = arow[kc + kk + kb + 1];
      v2f bb;
      bb.x = bt[kk + kb];
      bb.y = bt[kk + kb + 1];
      c = mma4(a, bb, c);
    }
    __syncthreads();
  }
  const int rb = (lane >> 4) * 8;
  const int l = lblk + wave * 16 + m;
#pragma unroll
  for (int r = 0; r < 8; ++r)
    logits[((long)b * H_ + r + rb) * (long)L + l] = c[r] * scale;
}

// ---------------------------------------------------------------------------
// ctx[b,h,d] = sum_l attn[b,h,l] * kv[b,l,d]
// Block tiles 64 d-columns; 64 l-rows per chunk staged through LDS (async,
// double-buffered). B frag: lanes 0-15 read one LDS row contiguously.
// ---------------------------------------------------------------------------
__global__ __launch_bounds__(128) void ctx_kernel(
    const float* __restrict__ attn,  // [B,H,L]
    const float* __restrict__ hist,  // [B,LP,D]
    const float* __restrict__ last,  // [B,D]
    float* __restrict__ ctx,         // [B,H,D]
    int LP, int L) {
  __shared__ float tile[2 * TILEF_];
  const int b = blockIdx.y;
  const int t = threadIdx.x;
  const int lane = t & 31;
  const int wave = t >> 5;
  const int dblk = blockIdx.x * 64;
  const int m = lane & 15;
  const int kb = (lane >> 4) << 1;

  const int lrow = t >> 1;        // l-row within chunk
  const int lcol = (t & 1) * 32;  // d-column offset within 64-wide tile

  const float* arow = attn + ((long)b * H_ + m) * (long)L;
  const int dloc = wave * 16 + m;

  // preload chunk 0 (rows 0..63)
  {
    const float* g = ((lrow < LP) ? (hist + ((long)b * LP + lrow) * D_)
                                  : (last + (long)b * D_)) +
                     dblk + lcol;
    float* dst = &tile[lrow * STRIDE_ + lcol];
#pragma unroll
    for (int i = 0; i < 8; ++i) async_cp16(dst + i * 4, g + i * 4);
  }

  v8f c = {};
  for (int lc = 0; lc < L; lc += ROWS_) {
    const int cur = (lc / ROWS_) & 1;
    if (lc + ROWS_ < L) {
      const int lg = lc + ROWS_ + lrow;
      const float* g = ((lg < LP) ? (hist + ((long)b * LP + lg) * D_)
                                  : (last + (long)b * D_)) +
                       dblk + lcol;
      float* dst = &tile[(cur ^ 1) * TILEF_ + lrow * STRIDE_ + lcol];
#pragma unroll
      for (int i = 0; i < 8; ++i) async_cp16(dst + i * 4, g + i * 4);
      ASYNC_WAIT(8);
    } else {
      ASYNC_WAIT(0);
    }
    __syncthreads();
    const float* bt = &tile[cur * TILEF_];
#pragma unroll
    for (int ll = 0; ll < ROWS_; ll += 4) {
      v2f a;
      a.x = arow[lc + ll + kb];
      a.y = arow[lc + ll + kb + 1];
      v2f bb;
      bb.x = bt[(ll + kb) * STRIDE_ + dloc];
      bb.y = bt[(ll + kb + 1) * STRIDE_ + dloc];
      c = mma4(a, bb, c);
    }
    __syncthreads();
  }
  const int rb = (lane >> 4) * 8;
#pragma unroll
  for (int r = 0; r < 8; ++r)
    ctx[((long)b * H_ + r + rb) * D_ + dblk + wave * 16 + m] = c[r];
}

// ---------------------------------------------------------------------------
// Row softmax over L; one block (256 threads) per (b,h) row.
// Writes normalized attention into both the ws copy and the d_out region.
// ---------------------------------------------------------------------------
__global__ __launch_bounds__(256) void softmax_kernel(
    const float* __restrict__ logits, float* __restrict__ attn_ws,
    float* __restrict__ attn_out, int L) {
  const int row = blockIdx.x;  // b*H + h
  const float* x = logits + (long)row * L;
  __shared__ float redA[8];
  __shared__ float redB[8];
  const int t = threadIdx.x;
  float mx = -3.4e38f;
  for (int i = t; i < L; i += 256) mx = fmaxf(mx, x[i]);
#pragma unroll
  for (int o = 16; o; o >>= 1) mx = fmaxf(mx, __shfl_xor(mx, o, 32));
  if ((t & 31) == 0) redA[t >> 5] = mx;
  __syncthreads();
  mx = redA[0];
#pragma unroll
  for (int i = 1; i < 8; ++i) mx = fmaxf(mx, redA[i]);
  float s = 0.0f;
  for (int i = t; i < L; i += 256) s += __expf(x[i] - mx);
#pragma unroll
  for (int o = 16; o; o >>= 1) s += __shfl_xor(s, o, 32);
  if ((t & 31) == 0) redB[t >> 5] = s;
  __syncthreads();
  s = 0.0f;
#pragma unroll
  for (int i = 0; i < 8; ++i) s += redB[i];
  const float inv = 1.0f / s;
  for (int i = t; i < L; i += 256) {
    const float v = __expf(x[i] - mx) * inv;
    attn_ws[(long)row * L + i] = v;
    attn_out[(long)row * L + i] = v;
  }
}

// ---------------------------------------------------------------------------
// out[b,:] = LayerNorm(a[b,:] + r[b,:]) * g + beta     (D=1024, 256 thr/blk)
// ---------------------------------------------------------------------------
__global__ __launch_bounds__(256) void add_ln_kernel(
    const float* __restrict__ a, const float* __restrict__ r,
    const float* __restrict__ g, const float* __restrict__ beta,
    float* __restrict__ out) {
  const int b = blockIdx.x;
  const int t = threadIdx.x;
  __shared__ float red[8];
  float local[4];
  float s = 0.0f;
#pragma unroll
  for (int i = 0; i < 4; ++i) {
    const int idx = t + i * 256;
    local[i] = a[(long)b * D_ + idx] + r[(long)b * D_ + idx];
    s += local[i];
  }
#pragma unroll
  for (int o = 16; o; o >>= 1) s += __shfl_xor(s, o, 32);
  if ((t & 31) == 0) red[t >> 5] = s;
  __syncthreads();
  s = 0.0f;
#pragma unroll
  for (int i = 0; i < 8; ++i) s += red[i];
  const float mean = s * (1.0f / D_);
  __syncthreads();
  float v = 0.0f;
#pragma unroll
  for (int i = 0; i < 4; ++i) {
    const float d = local[i] - mean;
    v += d * d;
  }
#pragma unroll
  for (int o = 16; o; o >>= 1) v += __shfl_xor(v, o, 32);
  if ((t & 31) == 0) red[t >> 5] = v;
  __syncthreads();
  v = 0.0f;
#pragma unroll
  for (int i = 0; i < 8; ++i) v += red[i];
  const float inv = rsqrtf(v * (1.0f / D_) + 1e-5f);
#pragma unroll
  for (int i = 0; i < 4; ++i) {
    const int idx = t + i * 256;
    out[(long)b * D_ + idx] = (local[i] - mean) * inv * g[idx] + beta[idx];
  }
}

// ---------------------------------------------------------------------------
extern "C" void kernel_launch(void* const* d_in, const int* in_sizes, int n_in,
                              void* d_out, int out_size, void* d_ws,
                              size_t ws_size, hipStream_t stream) {
  const float* dec    = (const float*)d_in[0];   // [B,1,D]
  const float* hist   = (const float*)d_in[1];   // [B,LP,D]
  const float* enc    = (const float*)d_in[2];   // [B,LE,D]
  const float* wq_s   = (const float*)d_in[3];   // [H,D,DK]
  const float* wk_s   = (const float*)d_in[4];
  const float* wv_s   = (const float*)d_in[5];
  const float* wo_s   = (const float*)d_in[6];   // [H*DV, D]
  const float* bo_s   = (const float*)d_in[7];
  const float* ln1_g  = (const float*)d_in[8];
  const float* ln1_b  = (const float*)d_in[9];
  const float* wq_c   = (const float*)d_in[10];
  const float* wk_c   = (const float*)d_in[11];
  const float* wv_c   = (const float*)d_in[12];
  const float* wo_c   = (const float*)d_in[13];
  const float* bo_c   = (const float*)d_in[14];
  const float* ln2_g  = (const float*)d_in[15];
  const float* ln2_b  = (const float*)d_in[16];
  const float* ffn_w1 = (const float*)d_in[17];  // [D,DF]
  const float* ffn_b1 = (const float*)d_in[18];
  const float* ffn_w2 = (const float*)d_in[19];  // [DF,D]
  const float* ffn_b2 = (const float*)d_in[20];
  const float* ln3_g  = (const float*)d_in[21];
  const float* ln3_b  = (const float*)d_in[22];

  float* out  = (float*)d_out;                 // [B,1,D]
  float* slfA = out + B_ * D_;                 // [B,H,1,LS]
  float* encA = slfA + (long)B_ * H_ * LSC;    // [B,H,1,LE]

  float* ws     = (float*)d_ws;
  float* q      = ws;                           // B*H*DK
  float* qw     = q + B_ * H_ * DK_;            // B*H*D
  float* logits = qw + B_ * H_ * D_;            // B*H*2048
  float* attn   = logits + (long)B_ * H_ * LSC; // B*H*2048
  float* ctx    = attn + (long)B_ * H_ * LSC;   // B*H*D
  float* ohead  = ctx + B_ * H_ * D_;           // B*H*DV
  float* oproj  = ohead + B_ * H_ * DV_;        // B*D
  float* x1     = oproj + B_ * D_;              // B*D
  float* x2     = x1 + B_ * D_;                 // B*D
  float* ffnh   = x2 + B_ * D_;                 // B*DF
  float* y      = ffnh + B_ * DF_;              // B*D

  const dim3 blk(128);
  const float scale = 0.125f;  // 1/sqrt(DK)

  // ===================== self attention =====================
  gemm16_kernel<0, 0><<<dim3(1, H_), blk, 0, stream>>>(
      dec, D_, 0L, wq_s, DK_, (long)D_ * DK_, nullptr, 0L,
      q, H_ * DK_, (long)DK_, D_);
  gemm16_kernel<1, 0><<<dim3(D_ / 64, H_), blk, 0, stream>>>(
      q, H_ * DK_, (long)DK_, wk_s, DK_, (long)D_ * DK_, nullptr, 0L,
      qw, H_ * D_, (long)D_, DK_);
  scores_kernel<<<dim3(LSC / 64, B_), blk, 0, stream>>>(
      qw, hist, dec, logits, LPC, LSC, scale);
  softmax_kernel<<<dim3(B_ * H_), dim3(256), 0, stream>>>(
      logits, attn, slfA, LSC);
  ctx_kernel<<<dim3(D_ / 64, B_), blk, 0, stream>>>(
      attn, hist, dec, ctx, LPC, LSC);
  gemm16_kernel<0, 0><<<dim3(1, H_), blk, 0, stream>>>(
      ctx, H_ * D_, (long)D_, wv_s, DV_, (long)D_ * DV_, nullptr, 0L,
      ohead, H_ * DV_, (long)DV_, D_);
  gemm16_kernel<0, 0><<<dim3(D_ / 64, 1), blk, 0, stream>>>(
      ohead, H_ * DV_, 0L, wo_s, D_, 0L, bo_s, 0L,
      oproj, D_, 0L, H_ * DV_);
  add_ln_kernel<<<dim3(B_), dim3(256), 0, stream>>>(
      oproj, dec, ln1_g, ln1_b, x1);

  // ===================== cross attention =====================
  gemm16_kernel<0, 0><<<dim3(1, H_), blk, 0, stream>>>(
      x1, D_, 0L, wq_c, DK_, (long)D_ * DK_, nullptr, 0L,
      q, H_ * DK_, (long)DK_, D_);
  gemm16_kernel<1, 0><<<dim3(D_ / 64, H_), blk, 0, stream>>>(
      q, H_ * DK_, (long)DK_, wk_c, DK_, (long)D_ * DK_, nullptr, 0L,
      qw, H_ * D_, (long)D_, DK_);
  scores_kernel<<<dim3(LEC / 64, B_), blk, 0, stream>>>(
      qw, enc, enc, logits, LEC, LEC, scale);
  softmax_kernel<<<dim3(B_ * H_), dim3(256), 0, stream>>>(
      logits, attn, encA, LEC);
  ctx_kernel<<<dim3(D_ / 64, B_), blk, 0, stream>>>(
      attn, enc, enc, ctx, LEC, LEC);
  gemm16_kernel<0, 0><<<dim3(1, H_), blk, 0, stream>>>(
      ctx, H_ * D_, (long)D_, wv_c, DV_, (long)D_ * DV_, nullptr, 0L,
      ohead, H_ * DV_, (long)DV_, D_);
  gemm16_kernel<0, 0><<<dim3(D_ / 64, 1), blk, 0, stream>>>(
      ohead, H_ * DV_, 0L, wo_c, D_, 0L, bo_c, 0L,
      oproj, D_, 0L, H_ * DV_);
  add_ln_kernel<<<dim3(B_), dim3(256), 0, stream>>>(
      oproj, x1, ln2_g, ln2_b, x2);

  // ===================== FFN + final LN =====================
  gemm16_kernel<0, 1><<<dim3(DF_ / 64, 1), blk, 0, stream>>>(
      x2, D_, 0L, ffn_w1, DF_, 0L, ffn_b1, 0L,
      ffnh, DF_, 0L, D_);
  gemm16_kernel<0, 0><<<dim3(D_ / 64, 1), blk, 0, stream>>>(
      ffnh, DF_, 0L, ffn_w2, D_, 0L, ffn_b2, 0L,
      y, D_, 0L, DF_);
  add_ln_kernel<<<dim3(B_), dim3(256), 0, stream>>>(
      y, x2, ln3_g, ln3_b, out);
}